// CausalDepthwiseConv_10428180594722
// MI455X (gfx1250) — compile-verified
//
#include <hip/hip_runtime.h>
#include <hip/hip_bf16.h>

// Causal depthwise conv1d, (B,L,D)=(8,4096,1024), K=15, fp32.
// Memory-bound: ~268 MB moved -> ~11.5us at 23.3 TB/s. Strategy: stream x
// through LDS via CDNA5 async global->LDS B128 loads (halo shared per block),
// compute with register-resident weights + sliding register window, B128 stores.

#define DI 1024
#define LI 4096
#define BI 8
#define KI 15
#define HALO (KI - 1)        // 14
#define TL 64                // output time-rows per block
#define ROWS (TL + HALO)     // 78 rows staged in LDS
#define CH 128               // channels per block
#define CH4 (CH / 4)         // 32 float4 columns
#define RPT 8                // rows per thread
#define NTHREADS 256         // 8 waves (wave32)

struct alignas(16) f4 { float x, y, z, w; };

__global__ __launch_bounds__(NTHREADS)
void CausalDepthwiseConv_kernel(const float* __restrict__ x,
                                const float* __restrict__ w,
                                float* __restrict__ out)
{
    // 78 * 32 * 16B = 39,936 bytes of LDS
    __shared__ f4 tile[ROWS * CH4];

    const int tid  = threadIdx.x;
    const int b    = blockIdx.z;
    const int dblk = blockIdx.y * CH;   // channel-block base
    const int t0   = blockIdx.x * TL;   // first output row of this block

    // ---------------- stage input tile (with left halo) into LDS ----------------
    // Each lane issues async B128 copies global->LDS (tracked by ASYNCcnt).
    const float* xb = x + ((size_t)b * LI) * DI + dblk;
    for (int c = tid; c < ROWS * CH4; c += NTHREADS) {
        const int row = c >> 5;          // c / CH4
        const int col = c & (CH4 - 1);   // c % CH4
        const int gt  = t0 - HALO + row; // global time of this LDS row
        unsigned lds_addr = (unsigned)(unsigned long long)(const void*)&tile[c];
        if (gt >= 0) {
            const float* gp = xb + (size_t)gt * DI + col * 4;
            asm volatile("global_load_async_to_lds_b128 %0, %1, off"
                         :: "v"(lds_addr), "v"(gp)
                         : "memory");
        } else {
            // causal zero padding (only blocks at t0 == 0); normal DS store,
            // compiler orders it against the barrier via DScnt.
            tile[c] = f4{0.f, 0.f, 0.f, 0.f};
        }
    }

    // ---------------- per-thread weights: 4 channels x 15 taps -----------------
    const int col  = tid & (CH4 - 1);   // which float4 column (0..31)
    const int rseg = tid >> 5;          // which 8-row segment (0..7)
    const int d0   = dblk + col * 4;

    f4 wv[KI];
    #pragma unroll
    for (int k = 0; k < KI; ++k)
        wv[k] = *(const f4*)(w + (size_t)k * DI + d0);  // weight[k,0,d]

    // Compiler doesn't track ASYNCcnt for the inline asm above: drain it, then
    // barrier so every wave sees the whole staged tile.
    asm volatile("s_wait_asynccnt 0x0" ::: "memory");
    __syncthreads();

    // ---------------- compute: 22-row register window -> 8 outputs -------------
    const int rbase = rseg * RPT;
    f4 win[RPT + HALO];                 // 22 float4 = 88 VGPRs
    #pragma unroll
    for (int i = 0; i < RPT + HALO; ++i)
        win[i] = tile[(rbase + i) * CH4 + col];

    // out[t] = sum_k w[k] * x[t - 14 + k]; LDS row j holds time t0 - 14 + j,
    // so output row (rbase + r) consumes win[r .. r+14].
    float* ob = out + (((size_t)b * LI) + t0 + rbase) * DI + d0;
    #pragma unroll
    for (int r = 0; r < RPT; ++r) {
        f4 acc = {0.f, 0.f, 0.f, 0.f};
        #pragma unroll
        for (int k = 0; k < KI; ++k) {
            const f4 a  = win[r + k];
            const f4 ww = wv[k];
            acc.x = fmaf(ww.x, a.x, acc.x);
            acc.y = fmaf(ww.y, a.y, acc.y);
            acc.z = fmaf(ww.z, a.z, acc.z);
            acc.w = fmaf(ww.w, a.w, acc.w);
        }
        *(f4*)(ob + (size_t)r * DI) = acc;   // B128 store
    }
}

extern "C" void kernel_launch(void* const* d_in, const int* in_sizes, int n_in,
                              void* d_out, int out_size, void* d_ws, size_t ws_size,
                              hipStream_t stream)
{
    const float* x   = (const float*)d_in[0];   // (8, 4096, 1024) fp32
    const float* wgt = (const float*)d_in[1];   // (15, 1, 1024)   fp32
    float* out       = (float*)d_out;           // (8, 4096, 1024) fp32

    dim3 grid(LI / TL, DI / CH, BI);            // (64, 8, 8) = 4096 blocks
    dim3 block(NTHREADS);
    CausalDepthwiseConv_kernel<<<grid, block, 0, stream>>>(x, wgt, out);
}